// EdgeModel_35012573397501
// MI455X (gfx1250) — compile-verified
//
#include <hip/hip_runtime.h>

#define HIDDEN 128

typedef __attribute__((ext_vector_type(2))) float v2f;
typedef __attribute__((ext_vector_type(8))) float v8f;

// ---------------------------------------------------------------------------
// Phase 1: per-node projections via V_WMMA_F32_16X16X4_F32.
//   s[n] = x[n] . W[:,0:128]
//   t[n] = x[n] . W[:,128:256] + b     (bias folded in here)
// One wave handles one 16-node tile; K=128 consumed in 32 steps of 4.
// B-matrix: column 0 = W_src, column 1 = W_dst, columns 2..15 zero.
// ---------------------------------------------------------------------------
__global__ __launch_bounds__(256)
void edgemodel_node_proj(const float* __restrict__ x,
                         const float* __restrict__ W,
                         const float* __restrict__ b,
                         float* __restrict__ s,
                         float* __restrict__ t,
                         int n_nodes, int n_tiles)
{
    __shared__ float sW[2 * HIDDEN];
    const int tid = threadIdx.x;
    if (tid < 2 * HIDDEN) sW[tid] = W[tid];
    __syncthreads();

    const int wave = tid >> 5;          // wave32 on gfx1250
    const int lane = tid & 31;
    const int tile = blockIdx.x * 8 + wave;
    if (tile >= n_tiles) return;        // wave-uniform: EXEC stays all-1s below

    const int node0 = tile * 16;
    const int nsub  = lane & 15;        // M index for A, N index for B
    const int half  = lane >> 4;        // selects K pair: 0 -> K{0,1}, 1 -> K{2,3}
    const int kb    = half * 2;

    // Clamp row so OOB tiles (if n_nodes not multiple of 16) read valid memory.
    int row = node0 + nsub;
    if (row >= n_nodes) row = n_nodes - 1;
    const float* __restrict__ xrow = x + (size_t)row * HIDDEN;

    // Lane's B column: 0 -> W_src, 1 -> W_dst, else zero.
    const float wscale = (nsub < 2) ? 1.0f : 0.0f;
    const int   wbase  = (nsub == 1) ? HIDDEN : 0;

    v8f acc = {0.f, 0.f, 0.f, 0.f, 0.f, 0.f, 0.f, 0.f};

    #pragma unroll
    for (int k0 = 0; k0 < HIDDEN; k0 += 4) {
        const int k = k0 + kb;
        v2f a;
        a.x = xrow[k];
        a.y = xrow[k + 1];
        v2f bv;
        bv.x = sW[wbase + k]     * wscale;
        bv.y = sW[wbase + k + 1] * wscale;
        // D = A(16x4) * B(4x16) + C ; 8-arg form:
        // (neg_a, A, neg_b, B, c_mod, C, reuse_a, reuse_b)
        acc = __builtin_amdgcn_wmma_f32_16x16x4_f32(
            false, a, false, bv, (short)0, acc, false, false);
    }

    // D layout: lanes 0-15 hold N=lane with M = vgpr(0..7);
    //           lanes 16-31 hold N=lane-16 with M = vgpr+8.
    const float bias = b[0];
    if (nsub == 0) {                         // column 0 -> s
        int base = node0 + half * 8;
        float* __restrict__ p = s + base;
        #pragma unroll
        for (int r = 0; r < 8; ++r)
            if (base + r < n_nodes) p[r] = acc[r];
    } else if (nsub == 1) {                  // column 1 -> t (+bias)
        int base = node0 + half * 8;
        float* __restrict__ p = t + base;
        #pragma unroll
        for (int r = 0; r < 8; ++r)
            if (base + r < n_nodes) p[r] = acc[r] + bias;
    }
}

// ---------------------------------------------------------------------------
// Phase 2: out[e] = s[src[e]] + t[dst[e]].  s/t (800 KB) live in L2.
// Two edges per thread; 128-bit loads of the int64 indices.
// ---------------------------------------------------------------------------
struct LL2 { long long x, y; };

__global__ __launch_bounds__(256)
void edgemodel_edge_gather(const long long* __restrict__ ei,
                           const float* __restrict__ s,
                           const float* __restrict__ t,
                           float* __restrict__ out,
                           int n_edges)
{
    const int i = blockIdx.x * blockDim.x + threadIdx.x;
    const int e = i * 2;
    if (e + 1 < n_edges) {
        LL2 sp = *(const LL2*)(ei + e);             // src pair
        LL2 dp = *(const LL2*)(ei + n_edges + e);   // dst pair
        float2 r;
        r.x = s[sp.x] + t[dp.x];
        r.y = s[sp.y] + t[dp.y];
        *(float2*)(out + e) = r;
    } else if (e < n_edges) {
        long long si = ei[e];
        long long di = ei[n_edges + e];
        out[e] = s[si] + t[di];
    }
}

// ---------------------------------------------------------------------------
// Inputs (setup_inputs order): 0=x (f32, 100000*128), 1=edge_index (i64, 2*E),
// 2=W (f32, 256), 3=b (f32, 1).  Output: E floats.
// Workspace: s[n_nodes] then t[n_nodes] floats (~800 KB).
// ---------------------------------------------------------------------------
extern "C" void kernel_launch(void* const* d_in, const int* in_sizes, int n_in,
                              void* d_out, int out_size, void* d_ws, size_t ws_size,
                              hipStream_t stream)
{
    const float*     x  = (const float*)d_in[0];
    const long long* ei = (const long long*)d_in[1];
    const float*     W  = (const float*)d_in[2];
    const float*     b  = (const float*)d_in[3];
    float*           out = (float*)d_out;

    const int n_nodes = in_sizes[0] / HIDDEN;
    const int n_edges = in_sizes[1] / 2;

    float* s = (float*)d_ws;
    float* t = s + n_nodes;

    const int n_tiles = (n_nodes + 15) / 16;
    const int blocks1 = (n_tiles + 7) / 8;          // 8 waves (tiles) per block
    edgemodel_node_proj<<<blocks1, 256, 0, stream>>>(x, W, b, s, t, n_nodes, n_tiles);

    const int pairs   = (n_edges + 1) / 2;
    const int blocks2 = (pairs + 255) / 256;
    edgemodel_edge_gather<<<blocks2, 256, 0, stream>>>(ei, s, t, out, n_edges);
}